// GCLayer_57896159150659
// MI455X (gfx1250) — compile-verified
//
#include <hip/hip_runtime.h>

typedef __attribute__((ext_vector_type(2))) float v2f;
typedef __attribute__((ext_vector_type(8))) float v8f;

#define IN_F 128
#define OUT_F 64
#define WAVES_PER_BLOCK 8

// ---------------------------------------------------------------------------
// Kernel 1: out[n][f] = bias[f]   (folds the +bias in exactly once, and
// initializes the poisoned d_out before the scatter-add atomics)
// ---------------------------------------------------------------------------
__global__ void init_out_kernel(float* __restrict__ out,
                                const float* __restrict__ bias, int total) {
  int i = blockIdx.x * blockDim.x + threadIdx.x;
  if (i < total) out[i] = bias[i & (OUT_F - 1)];
}

// ---------------------------------------------------------------------------
// Kernel 2: support = vertex @ W via V_WMMA_F32_16X16X4_F32 (f32, wave32).
// One wave computes a full 16x64 output tile: each A (v2f) load is reused by
// 4 WMMAs -> vertex is read exactly once (the 51 MB bandwidth floor), and the
// 4 accumulator chains are independent so WMMAs pipeline back-to-back.
//
// W is staged into LDS interleaved by K-pairs: wlds2[p][n] = {W[2p][n],
// W[2p+1][n]}. The B operand of each WMMA (two K values at one column) is
// then a single contiguous ds_load_b64 into an adjacent VGPR pair — no
// register shuffles between LDS load and v_wmma.
//
// A 16x4 f32 layout: lanes 0-15 hold K=0,1 ; lanes 16-31 hold K=2,3.
// ---------------------------------------------------------------------------
__global__ __launch_bounds__(256) void gemm_wmma_kernel(
    const float* __restrict__ V, const float* __restrict__ W,
    float* __restrict__ S, int nRowTiles) {
  __shared__ v2f wlds2[(IN_F / 2) * OUT_F];  // 32 KB of the 320 KB WGP LDS
  for (int i = threadIdx.x; i < (IN_F / 2) * OUT_F; i += blockDim.x) {
    const int p = i / OUT_F;  // K-pair index
    const int n = i % OUT_F;  // output column
    v2f w;
    w.x = W[(2 * p + 0) * OUT_F + n];
    w.y = W[(2 * p + 1) * OUT_F + n];
    wlds2[i] = w;
  }
  __syncthreads();

  const int lane = threadIdx.x & 31;
  const int wave = threadIdx.x >> 5;
  const int rowTile = blockIdx.x * WAVES_PER_BLOCK + wave;
  if (rowTile >= nRowTiles) return;  // wave-uniform: EXEC all-1s for WMMA

  const int m    = lane & 15;         // M (for A) / N (for B) index
  const int koff = (lane >> 4) << 1;  // 0 for lanes 0-15, 2 for lanes 16-31

  const float* vptr = V + (long long)(rowTile * 16 + m) * IN_F + koff;
  const v2f*   wbase = wlds2 + (koff >> 1) * OUT_F + m;

  v8f c0 = {}, c1 = {}, c2 = {}, c3 = {};
#pragma unroll
  for (int k = 0; k < IN_F; k += 4) {
    v2f a = *(const v2f*)(vptr + k);       // A[m][k+koff .. k+koff+1]
    const v2f* wrow = wbase + (k >> 1) * OUT_F;  // pair-row (k+koff)/2
    v2f b0 = wrow[0];                      // cols m, m+16, m+32, m+48
    v2f b1 = wrow[16];
    v2f b2 = wrow[32];
    v2f b3 = wrow[48];
    // 8 args: (neg_a, A, neg_b, B, c_mod, C, reuse_a, reuse_b)
    c0 = __builtin_amdgcn_wmma_f32_16x16x4_f32(false, a, false, b0,
                                               (short)0, c0, false, false);
    c1 = __builtin_amdgcn_wmma_f32_16x16x4_f32(false, a, false, b1,
                                               (short)0, c1, false, false);
    c2 = __builtin_amdgcn_wmma_f32_16x16x4_f32(false, a, false, b2,
                                               (short)0, c2, false, false);
    c3 = __builtin_amdgcn_wmma_f32_16x16x4_f32(false, a, false, b3,
                                               (short)0, c3, false, false);
  }

  // C/D f32 16x16: VGPR r holds row M=r (lanes 0-15) / M=r+8 (lanes 16-31)
  const long long mBase = (long long)rowTile * 16 + ((lane >> 4) << 3);
#pragma unroll
  for (int r = 0; r < 8; ++r) {
    float* srow = S + (mBase + r) * OUT_F + m;
    srow[0]  = c0[r];
    srow[16] = c1[r];
    srow[32] = c2[r];
    srow[48] = c3[r];
  }
}

// ---------------------------------------------------------------------------
// Kernel 3: COO SpMM scatter-add. 16 threads per edge, float4 gather of
// support[col], 4x global f32 atomic adds into out[row] (L2 atomic units;
// whole 25.6 MB output fits in the 192 MB L2).
// ---------------------------------------------------------------------------
__global__ void spmm_kernel(const float* __restrict__ S,
                            const int* __restrict__ erow,
                            const int* __restrict__ ecol,
                            const float* __restrict__ eval,
                            float* out, int nWork) {
  int idx = blockIdx.x * blockDim.x + threadIdx.x;
  if (idx >= nWork) return;
  const int e = idx >> 4;          // edge id
  const int f = (idx & 15) << 2;   // feature quad
  const int col = ecol[e];
  const int row = erow[e];
  const float v = eval[e];
  const float4 s = *(const float4*)(S + (long long)col * OUT_F + f);
  float* o = out + (long long)row * OUT_F + f;
  atomicAdd(o + 0, s.x * v);
  atomicAdd(o + 1, s.y * v);
  atomicAdd(o + 2, s.z * v);
  atomicAdd(o + 3, s.w * v);
}

// ---------------------------------------------------------------------------
extern "C" void kernel_launch(void* const* d_in, const int* in_sizes, int n_in,
                              void* d_out, int out_size, void* d_ws,
                              size_t ws_size, hipStream_t stream) {
  const float* vertex  = (const float*)d_in[0];
  const int*   erow    = (const int*)d_in[1];
  const int*   ecol    = (const int*)d_in[2];
  const float* eval    = (const float*)d_in[3];
  const float* weights = (const float*)d_in[4];
  const float* bias    = (const float*)d_in[5];
  float* out     = (float*)d_out;
  float* support = (float*)d_ws;  // nNodes * OUT_F floats (25.6 MB)

  const int nNodes = in_sizes[0] / IN_F;  // 100000 (divisible by 16)
  const int nEdges = in_sizes[1];         // 1600000

  const int totalOut = nNodes * OUT_F;
  init_out_kernel<<<(totalOut + 255) / 256, 256, 0, stream>>>(out, bias,
                                                              totalOut);

  const int nRowTiles = nNodes / 16;  // 6250
  const int nBlocks = (nRowTiles + WAVES_PER_BLOCK - 1) / WAVES_PER_BLOCK;
  gemm_wmma_kernel<<<nBlocks, 256, 0, stream>>>(vertex, weights, support,
                                                nRowTiles);

  const int nWork = nEdges * 16;
  spmm_kernel<<<(nWork + 255) / 256, 256, 0, stream>>>(support, erow, ecol,
                                                       eval, out, nWork);
}